// Collaborative_Filtering_13675175870752
// MI455X (gfx1250) — compile-verified
//
#include <hip/hip_runtime.h>

typedef __attribute__((ext_vector_type(2))) float v2f;
typedef __attribute__((ext_vector_type(4))) float v4f;
typedef __attribute__((ext_vector_type(8))) float v8f;

#define ACF_S     200
#define ACF_P     8
#define ACF_F     64
#define ACF_UPAD  66                  // F + 2: 8B alignment per row, no bank conflicts
#define ACF_RPAD  20                  // r-tile row stride (dwords): conflict-free stores+loads
#define ACF_TILES ((ACF_S + 15) / 16) // 13

__global__ __launch_bounds__(256) void acf_wmma_kernel(
    const int*   __restrict__ user,          // [B]
    const int*   __restrict__ items,         // [B, S]
    const float* __restrict__ user_factors,  // [N_USERS, P, F]
    const float* __restrict__ item_factors,  // [N_ITEMS, F]
    float*       __restrict__ pred_out,      // [B, S]
    float*       __restrict__ scores_out)    // [B, S, P]
{
    __shared__ float u_lds[ACF_P * ACF_UPAD];            // staged user factors
    __shared__ float r_lds[8][16 * ACF_RPAD];            // per-wave r-tile transpose buffer

    const int b    = blockIdx.x;
    const int tid  = threadIdx.x;
    const int lane = tid & 31;
    const int wave = tid >> 5;

    // ---- Stage u = user_factors[user[b]] (8x64 = 512 floats) into LDS ----
    const int uu = user[b];
    const float* up = user_factors + (size_t)uu * (ACF_P * ACF_F);
    for (int i = tid; i < ACF_P * ACF_F; i += 256) {
        u_lds[(i >> 6) * ACF_UPAD + (i & 63)] = up[i];
    }
    __syncthreads();

    const int half = lane >> 4;        // which half-wave
    const int lm   = lane & 15;        // A: item row within tile; B: persona column
    const int koff = half * 2;         // K offset within a chunk of 4 (ISA 16x4 f32 layout)

    // B-fragment source: persona row in LDS (zero-column for persona >= 8)
    const float bmask = (lm < ACF_P) ? 1.0f : 0.0f;
    const float* brow = u_lds + (lm & 7) * ACF_UPAD + koff;

    float* rrow = &r_lds[wave][0];

    for (int t = wave; t < ACF_TILES; t += 8) {
        const int s0 = t * 16;
        // clamp for tail tile (items 192..199 -> rows 8..15 duplicate row 199)
        const int sA = min(s0 + lm, ACF_S - 1);
        const int it = items[(size_t)b * ACF_S + sA];
        const float* vrow = item_factors + (size_t)it * ACF_F + koff;

        // ---- r[16 items x 8 personas] = v * u^T via 16 chained f32 WMMAs ----
        v8f acc = {};
#pragma unroll
        for (int kc = 0; kc < 16; ++kc) {
            const v2f a = *(const v2f*)(vrow + kc * 4);   // 8B-aligned global_load_b64
            v2f bf;
            bf.x = brow[kc * 4]     * bmask;
            bf.y = brow[kc * 4 + 1] * bmask;
            acc = __builtin_amdgcn_wmma_f32_16x16x4_f32(
                false, a, false, bf, (short)0, acc, false, false);
        }

        // ---- Transpose D tile through LDS ----
        // D layout: vgpr g, lane L -> r[item = g + 8*(L>>4)][persona = L&15].
        // Store all 16 "persona" columns (8..15 are garbage, never read back).
#pragma unroll
        for (int g = 0; g < 8; ++g) {
            rrow[(g + half * 8) * ACF_RPAD + lm] = acc[g];
        }
        // Cross-lane LDS dependency within this wave: one dscnt drain makes the
        // tile visible. Same-wave LDS ops are in-order, so reads below also
        // cannot be overtaken by next iteration's stores.
        asm volatile("s_wait_dscnt 0x0" ::: "memory");

        // ---- Per-item softmax over 8 personas + pred = <softmax(r), r> ----
        // Lanes 0..15 each own one item: 8 contiguous floats -> 2x ds_load_b128.
        const int s = s0 + lane;
        if (lane < 16 && s < ACF_S) {
            const float* rp = rrow + lane * ACF_RPAD;
            const v4f ra = *(const v4f*)(rp);       // personas 0..3
            const v4f rb = *(const v4f*)(rp + 4);   // personas 4..7

            float r8[8] = { ra.x, ra.y, ra.z, ra.w, rb.x, rb.y, rb.z, rb.w };

            float mx = r8[0];
#pragma unroll
            for (int p = 1; p < 8; ++p) mx = fmaxf(mx, r8[p]);

            float e8[8];
            float se = 0.0f, dr = 0.0f;
#pragma unroll
            for (int p = 0; p < 8; ++p) {
                const float e = __expf(r8[p] - mx);
                e8[p] = e;
                se += e;
                dr += e * r8[p];
            }
            const float inv = 1.0f / se;

            v4f sa, sb;
            sa.x = e8[0] * inv; sa.y = e8[1] * inv; sa.z = e8[2] * inv; sa.w = e8[3] * inv;
            sb.x = e8[4] * inv; sb.y = e8[5] * inv; sb.z = e8[6] * inv; sb.w = e8[7] * inv;

            float* sp = scores_out + ((size_t)b * ACF_S + s) * ACF_P;
            *(v4f*)(sp)     = sa;                    // global_store_b128
            *(v4f*)(sp + 4) = sb;                    // global_store_b128
            pred_out[(size_t)b * ACF_S + s] = dr * inv;
        }
    }
}

extern "C" void kernel_launch(void* const* d_in, const int* in_sizes, int n_in,
                              void* d_out, int out_size, void* d_ws, size_t ws_size,
                              hipStream_t stream) {
    const int*   user         = (const int*)  d_in[0];
    const int*   items        = (const int*)  d_in[1];
    const float* user_factors = (const float*)d_in[2];
    const float* item_factors = (const float*)d_in[3];

    const int B = in_sizes[0];                // 4096

    float* pred_out   = (float*)d_out;                 // [B, S]
    float* scores_out = pred_out + (size_t)B * ACF_S;  // [B, S, P]

    acf_wmma_kernel<<<B, 256, 0, stream>>>(user, items, user_factors, item_factors,
                                           pred_out, scores_out);
}